// SSAM_67199058313541
// MI455X (gfx1250) — compile-verified
//
#include <hip/hip_runtime.h>

// ---------------------------------------------------------------------------
// SSAM: out = x * sigmoid((x*rowsum(Wq)) * (x*rowsum(Wk)))
//            = x * sigmoid(x^2 * (rowsum(Wq)[c] * rowsum(Wk)[c]))
// S=4096, N=16, C=512 -> 65536 rows of 512 f32 channels.
// Phase 1: rowsum via V_WMMA_F32_16X16X4_F32 (W x ones), f32-exact.
// Phase 2: bandwidth-bound streaming apply (256 MB -> ~11 us @ 23.3 TB/s).
// ---------------------------------------------------------------------------

typedef __attribute__((ext_vector_type(2))) float v2f;
typedef __attribute__((ext_vector_type(8))) float v8f;

#define C_DIM 512
#define ROWS  (4096 * 16)   // S * N

// ---------------------------------------------------------------------------
// Phase 1: rowsum[c] = sum_j W[c,j] computed as W(16x4 tiles) x ones(4x16).
// One wave (32 threads) per 16-row tile. 32 tiles per weight, 2 weights.
// A-matrix f32 16x4 VGPR layout (ISA 7.12.2):
//   lanes 0-15 : M=lane,    VGPR0=K0, VGPR1=K1
//   lanes 16-31: M=lane-16, VGPR0=K2, VGPR1=K3
// D layout: VGPR r -> M=r (lanes 0-15), M=8+r (lanes 16-31); all N identical
// because B == ones, so lane 0 / lane 16 each hold 8 rowsums.
// ---------------------------------------------------------------------------
__global__ __launch_bounds__(32) void ssam_rowsum_wmma(
    const float* __restrict__ Wq,
    const float* __restrict__ Wk,
    float* __restrict__ ws)   // ws[0..511]=rowsum(Wq), ws[512..1023]=rowsum(Wk)
{
    const int tile   = blockIdx.x & 31;           // 0..31 (16-row tile)
    const bool is_k  = (blockIdx.x & 32) != 0;    // which weight
    const float* W   = is_k ? Wk : Wq;
    float* out       = ws + (is_k ? C_DIM : 0);

    const int lane  = threadIdx.x;                // wave32: 0..31
    const int m     = lane & 15;
    const int row   = tile * 16 + m;
    const int kbase = (lane >> 4) * 2;            // 0 (lanes 0-15) / 2 (16-31)
    const float* wrow = W + row * C_DIM + kbase;  // 8B-aligned

    v2f b; b[0] = 1.0f; b[1] = 1.0f;              // ones: layout-independent
    v8f c = {};

    for (int kk = 0; kk < C_DIM; kk += 4) {
        float2 t = *reinterpret_cast<const float2*>(wrow + kk);  // K, K+1
        v2f a; a[0] = t.x; a[1] = t.y;
        // 8 args: (neg_a, A, neg_b, B, c_mod, C, reuse_a, reuse_b)
        c = __builtin_amdgcn_wmma_f32_16x16x4_f32(
                false, a, false, b, (short)0, c, false, false);
    }

    // Extract column N=0 of D (all columns equal).
    if (lane == 0) {
        #pragma unroll
        for (int r = 0; r < 8; ++r) out[tile * 16 + r] = c[r];
    } else if (lane == 16) {
        #pragma unroll
        for (int r = 0; r < 8; ++r) out[tile * 16 + 8 + r] = c[r];
    }
}

// ---------------------------------------------------------------------------
// Phase 2: streaming apply. 128 threads/block = exactly one row of 512
// channels as float4 per thread -> fixed channels per thread, so the weight
// product wp = rowsum_q[c]*rowsum_k[c] lives in registers for the whole
// grid-stride loop. b128 loads/stores, rcp+exp TRANS sigmoid, prefetch ahead.
// ---------------------------------------------------------------------------
__device__ __forceinline__ float fast_sigmoid(float t) {
    // 1 / (1 + exp(-t)) : v_exp_f32 + v_rcp_f32 (TRANS, co-executes w/ VALU)
    return __builtin_amdgcn_rcpf(1.0f + __expf(-t));
}

__global__ __launch_bounds__(128) void ssam_apply(
    const float4* __restrict__ x4,
    const float*  __restrict__ ws,
    float4* __restrict__ o4,
    int nrows)
{
    const int tid = threadIdx.x;                          // 0..127
    const float4 wq = reinterpret_cast<const float4*>(ws)[tid];
    const float4 wk = reinterpret_cast<const float4*>(ws + C_DIM)[tid];
    const float wp0 = wq.x * wk.x;
    const float wp1 = wq.y * wk.y;
    const float wp2 = wq.z * wk.z;
    const float wp3 = wq.w * wk.w;

    const int vec_per_row = C_DIM / 4;                    // 128
    const int stride      = (int)gridDim.x * vec_per_row; // vecs per grid step
    const int end         = nrows * vec_per_row;          // 8.4M < 2^31
    int idx = (int)blockIdx.x * vec_per_row + tid;

    for (; idx < end; idx += stride) {
        // Prefetch ~4 iterations ahead (clamped in-bounds; speculative RT).
        int pidx = idx + 4 * stride;
        if (pidx >= end) pidx = idx;
        __builtin_prefetch(reinterpret_cast<const void*>(x4 + pidx), 0, 0);

        const float4 v = x4[idx];                         // global_load_b128
        float4 r;
        r.x = v.x * fast_sigmoid(v.x * v.x * wp0);
        r.y = v.y * fast_sigmoid(v.y * v.y * wp1);
        r.z = v.z * fast_sigmoid(v.z * v.z * wp2);
        r.w = v.w * fast_sigmoid(v.w * v.w * wp3);
        o4[idx] = r;                                      // global_store_b128
    }
}

// ---------------------------------------------------------------------------
extern "C" void kernel_launch(void* const* d_in, const int* in_sizes, int n_in,
                              void* d_out, int out_size, void* d_ws, size_t ws_size,
                              hipStream_t stream) {
    const float* x  = (const float*)d_in[0];   // (S,N,C)
    const float* Wq = (const float*)d_in[1];   // (1,1,C,C)
    const float* Wk = (const float*)d_in[2];   // (1,1,C,C)
    float* out = (float*)d_out;
    float* ws  = (float*)d_ws;                 // needs 1024 floats = 4 KB

    // Phase 1: 2 weights x 32 tiles, one wave32 each.
    ssam_rowsum_wmma<<<64, 32, 0, stream>>>(Wq, Wk, ws);

    // Phase 2: 4096 blocks x 128 threads -> 16 rows per block.
    ssam_apply<<<4096, 128, 0, stream>>>(
        reinterpret_cast<const float4*>(x), ws,
        reinterpret_cast<float4*>(out), ROWS);
}